// HinfNN_25237227832062
// MI455X (gfx1250) — compile-verified
//
#include <hip/hip_runtime.h>

#define NXc 128
#define NUc 64
#define NYc 32
#define KTOT 192      // 128 (x) + 64 (u)
#define COLS 160      // 128 (dx) + 32 (y)
#define LDST 196      // padded LDS row stride (floats); row start = 784B (16B aligned)
#define NS_ITERS 30
#define RT 4          // row tiles (of 16 samples) per block -> 64 samples/block

typedef __attribute__((ext_vector_type(2))) float v2f;
typedef __attribute__((ext_vector_type(8))) float v8f;

// ---------------------------------------------------------------------------
// Power iteration: scal[0] = 1.01 * sqrt(lambda_max(H H^T)),  H is 128x64
// ---------------------------------------------------------------------------
__global__ __launch_bounds__(128) void k_power(const float* __restrict__ H,
                                               float* __restrict__ scal) {
    __shared__ float v[128], w[64], red[128];
    int t = threadIdx.x;
    v[t] = 1.0f;
    __syncthreads();
    for (int it = 0; it < 50; ++it) {
        if (t < 64) {                       // w = H^T v
            float a = 0.f;
            for (int i = 0; i < 128; ++i) a += H[i * 64 + t] * v[i];
            w[t] = a;
        }
        __syncthreads();
        float a = 0.f;                      // v2 = H w
        for (int j = 0; j < 64; ++j) a += H[t * 64 + j] * w[j];
        red[t] = a * a;
        __syncthreads();
        for (int s = 64; s > 0; s >>= 1) {  // ||v2||^2
            if (t < s) red[t] += red[t + s];
            __syncthreads();
        }
        float invn = rsqrtf(red[0] + 1e-30f);
        __syncthreads();
        v[t] = a * invn;
        __syncthreads();
    }
    // Rayleigh quotient: lambda = ||H^T v||^2  (v normalized)
    if (t < 64) {
        float a = 0.f;
        for (int i = 0; i < 128; ++i) a += H[i * 64 + t] * v[i];
        w[t] = a;
    }
    __syncthreads();
    red[t] = (t < 64) ? w[t] * w[t] : 0.f;
    __syncthreads();
    for (int s = 64; s > 0; s >>= 1) {
        if (t < s) red[t] += red[t + s];
        __syncthreads();
    }
    if (t == 0) scal[0] = 1.01f * sqrtf(red[0]);
}

// ---------------------------------------------------------------------------
// Frobenius norm of Q, Newton-Schulz init: Y0 = Q/s, Z0 = I.
// scal[1] = s, scal[2] = sqrt(s)/denom   (B-matrix scale, gamma = 1)
// ---------------------------------------------------------------------------
__global__ __launch_bounds__(256) void k_frob_init(const float* __restrict__ Q,
                                                   float* __restrict__ scal,
                                                   float* __restrict__ Y,
                                                   float* __restrict__ Z) {
    __shared__ float red[256];
    int t = threadIdx.x;
    float a = 0.f;
    for (int i = t; i < NXc * NXc; i += 256) { float q = Q[i]; a += q * q; }
    red[t] = a;
    __syncthreads();
    for (int s = 128; s > 0; s >>= 1) {
        if (t < s) red[t] += red[t + s];
        __syncthreads();
    }
    float sf = sqrtf(red[0]);
    if (t == 0) { scal[1] = sf; scal[2] = sqrtf(sf) / scal[0]; }
    float inv = 1.0f / sf;
    for (int i = t; i < NXc * NXc; i += 256) {
        Y[i] = Q[i] * inv;
        Z[i] = ((i >> 7) == (i & 127)) ? 1.0f : 0.0f;
    }
}

// ---------------------------------------------------------------------------
// Generic small GEMM:  C = eff * op(A) @ B + identAdd * I
// ---------------------------------------------------------------------------
__global__ __launch_bounds__(256) void k_gemm(float* __restrict__ C,
                                              const float* __restrict__ A,
                                              const float* __restrict__ B,
                                              int M, int N, int K,
                                              int lda, int ldb, int ldc,
                                              float alpha, float identAdd,
                                              int transA,
                                              const float* __restrict__ scaleptr) {
    int id = blockIdx.x * blockDim.x + threadIdx.x;
    if (id >= M * N) return;
    int i = id / N, j = id % N;
    float acc = 0.f;
    if (transA) {
        for (int k = 0; k < K; ++k) acc += A[k * lda + i] * B[k * ldb + j];
    } else {
        for (int k = 0; k < K; ++k) acc += A[i * lda + k] * B[k * ldb + j];
    }
    float eff = alpha * (scaleptr ? *scaleptr : 1.0f);
    C[i * ldc + j] = eff * acc + ((i == j) ? identAdd : 0.0f);
}

// M1 = -0.5*(Q + G^T G + eps*I) + S
__global__ __launch_bounds__(256) void k_build_m1(float* __restrict__ M1,
                                                  const float* __restrict__ Q,
                                                  const float* __restrict__ GtG,
                                                  const float* __restrict__ S) {
    int id = blockIdx.x * blockDim.x + threadIdx.x;
    if (id >= NXc * NXc) return;
    int i = id >> 7, j = id & 127;
    float eps = (i == j) ? 1e-4f : 0.0f;
    M1[id] = -0.5f * (Q[id] + GtG[id] + eps) + S[id];
}

// ---------------------------------------------------------------------------
// Pack Wc (160x192) into the per-lane WMMA_F32_16X16X4 B-operand layout.
// ---------------------------------------------------------------------------
__global__ __launch_bounds__(256) void k_pack(float* __restrict__ Wpack,
                                              const float* __restrict__ A,
                                              const float* __restrict__ B,
                                              const float* __restrict__ C) {
    int id = blockIdx.x * blockDim.x + threadIdx.x;
    if (id >= (COLS / 16) * 48 * 32) return;
    int lane = id & 31;
    int kk4  = (id >> 5) % 48;
    int ct   = id / (48 * 32);
    int k0 = kk4 * 4 + ((lane >> 4) << 1);
    int c  = ct * 16 + (lane & 15);
    float v0, v1;
    if (c < NXc) {
        v0 = (k0     < NXc) ? A[c * NXc + k0]     : B[c * NUc + (k0     - NXc)];
        v1 = (k0 + 1 < NXc) ? A[c * NXc + k0 + 1] : B[c * NUc + (k0 + 1 - NXc)];
    } else {
        v0 = (k0     < NXc) ? C[(c - NXc) * NXc + k0]     : 0.0f;
        v1 = (k0 + 1 < NXc) ? C[(c - NXc) * NXc + k0 + 1] : 0.0f;
    }
    Wpack[id * 2 + 0] = v0;
    Wpack[id * 2 + 1] = v1;
}

// ---------------------------------------------------------------------------
// Main fused GEMM:  [dx | y](N x 160) = [x | u](N x 192) @ Wc^T
// Block: 320 threads (10 wave32's), 64 samples (4 row tiles). Each wave owns
// one 16-wide column tile and keeps 4 accumulators so every B-operand fetch
// is reused 4x (L2 weight traffic /4). Tiles staged to LDS with
// global_load_async_to_lds_b128 (ASYNCcnt) and fenced with s_wait_asynccnt.
// ---------------------------------------------------------------------------
__global__ __launch_bounds__(320) void k_main(const float* __restrict__ x,
                                              const float* __restrict__ u,
                                              const float* __restrict__ Wpack,
                                              float* __restrict__ out,
                                              long NN) {
    __shared__ float tile[RT * 16 * LDST];            // ~49 KB
    int t = threadIdx.x;
    long base = (long)blockIdx.x * (RT * 16);

    // ---- async stage: x (64x128) and u (64x64) as 16B chunks ----
    unsigned lbase = (unsigned)(size_t)(void*)tile;   // addr[31:0] = LDS offset
    {
        unsigned long gx = (unsigned long)(size_t)(x + base * NXc);
        const int XCH = RT * 16 * (NXc / 4);          // 2048 chunks
        for (int idx = t; idx < XCH; idx += 320) {
            int r  = idx >> 5;                        // 32 chunks per row
            int c4 = idx & 31;
            unsigned laddr = lbase + (unsigned)((r * LDST + c4 * 4) * 4);
            unsigned goff  = (unsigned)((r * NXc + c4 * 4) * 4);
            asm volatile("global_load_async_to_lds_b128 %0, %1, %2"
                         :: "v"(laddr), "v"(goff), "s"(gx) : "memory");
        }
        unsigned long gu = (unsigned long)(size_t)(u + base * NUc);
        const int UCH = RT * 16 * (NUc / 4);          // 1024 chunks
        for (int idx = t; idx < UCH; idx += 320) {
            int r  = idx >> 4;                        // 16 chunks per row
            int c4 = idx & 15;
            unsigned laddr = lbase + (unsigned)((r * LDST + NXc + c4 * 4) * 4);
            unsigned goff  = (unsigned)((r * NUc + c4 * 4) * 4);
            asm volatile("global_load_async_to_lds_b128 %0, %1, %2"
                         :: "v"(laddr), "v"(goff), "s"(gu) : "memory");
        }
    }
    asm volatile("s_wait_asynccnt 0x0" ::: "memory");
    __syncthreads();

    // ---- WMMA compute ----
    int wave = t >> 5, lane = t & 31;
    int half = lane >> 4, l16 = lane & 15;

    v8f acc[RT] = {};
    const v2f* wp = (const v2f*)Wpack;
    int wbase = wave * 48 * 32 + lane;
    const float* arow[RT];
#pragma unroll
    for (int rt = 0; rt < RT; ++rt) arow[rt] = &tile[(rt * 16 + l16) * LDST];

    for (int kk4 = 0; kk4 < 48; ++kk4) {
        int k0 = kk4 * 4 + half * 2;                  // A 16x4 f32 per-lane layout
        v2f b = wp[wbase + kk4 * 32];                 // one fetch, reused RT times
#pragma unroll
        for (int rt = 0; rt < RT; ++rt) {
            v2f a = *(const v2f*)(arow[rt] + k0);
            acc[rt] = __builtin_amdgcn_wmma_f32_16x16x4_f32(
                /*neg_a=*/false, a, /*neg_b=*/false, b,
                /*c_mod=*/(short)0, acc[rt], /*reuse_a=*/false, /*reuse_b=*/false);
        }
    }

    // ---- write out: VGPR r -> row r (lanes 0-15) / r+8 (lanes 16-31) ----
    if (wave < 8) {                                   // dx columns
        int col = wave * 16 + l16;
#pragma unroll
        for (int rt = 0; rt < RT; ++rt) {
            long rowb = base + rt * 16 + half * 8;
#pragma unroll
            for (int r = 0; r < 8; ++r)
                out[(rowb + r) * NXc + col] = acc[rt][r];
        }
    } else {                                          // y columns
        int col = (wave - 8) * 16 + l16;
        float* yo = out + NN * NXc;
#pragma unroll
        for (int rt = 0; rt < RT; ++rt) {
            long rowb = base + rt * 16 + half * 8;
#pragma unroll
            for (int r = 0; r < 8; ++r)
                yo[(rowb + r) * NYc + col] = acc[rt][r];
        }
    }
}

// ---------------------------------------------------------------------------
extern "C" void kernel_launch(void* const* d_in, const int* in_sizes, int n_in,
                              void* d_out, int out_size, void* d_ws, size_t ws_size,
                              hipStream_t stream) {
    const float* u = (const float*)d_in[0];
    const float* x = (const float*)d_in[1];
    const float* Q = (const float*)d_in[2];
    const float* P = (const float*)d_in[3];
    const float* S = (const float*)d_in[4];
    const float* G = (const float*)d_in[5];
    const float* H = (const float*)d_in[6];
    float* out = (float*)d_out;
    long NN = (long)in_sizes[1] / NXc;               // 262144

    // workspace layout (floats)
    float* ws    = (float*)d_ws;
    float* scal  = ws;                               // [0]=denom [1]=||Q||_F [2]=sqrt(s)/denom
    float* Y0    = ws + 16;
    float* Y1    = Y0 + NXc * NXc;
    float* Z0    = Y1 + NXc * NXc;
    float* Z1    = Z0 + NXc * NXc;
    float* Tm    = Z1 + NXc * NXc;
    float* GtG   = Tm + NXc * NXc;
    float* M1    = GtG + NXc * NXc;
    float* Amat  = M1 + NXc * NXc;
    float* Cmat  = Amat + NXc * NXc;                 // 32x128
    float* Bmat  = Cmat + NYc * NXc;                 // 128x64
    float* Wpack = Bmat + NXc * NUc;                 // 10*48*32*2 floats, 8B aligned

    auto gemm = [&](float* C, const float* A, const float* B, int M, int N, int K,
                    int lda, int ldb, int ldc, float alpha, float identAdd,
                    int transA, const float* scaleptr) {
        int total = M * N;
        k_gemm<<<(total + 255) / 256, 256, 0, stream>>>(
            C, A, B, M, N, K, lda, ldb, ldc, alpha, identAdd, transA, scaleptr);
    };

    // 1) spectral norm of H H^T -> denom
    k_power<<<1, 128, 0, stream>>>(H, scal);
    // 2) ||Q||_F, Newton-Schulz init
    k_frob_init<<<1, 256, 0, stream>>>(Q, scal, Y0, Z0);

    // 3) Newton-Schulz: T = 3I - Z Y; Y <- 0.5 Y T; Z <- 0.5 T Z
    float *Yc = Y0, *Yn = Y1, *Zc = Z0, *Zn = Z1;
    for (int it = 0; it < NS_ITERS; ++it) {
        gemm(Tm, Zc, Yc, NXc, NXc, NXc, NXc, NXc, NXc, -1.0f, 3.0f, 0, nullptr);
        gemm(Yn, Yc, Tm, NXc, NXc, NXc, NXc, NXc, NXc, 0.5f, 0.0f, 0, nullptr);
        gemm(Zn, Tm, Zc, NXc, NXc, NXc, NXc, NXc, NXc, 0.5f, 0.0f, 0, nullptr);
        float* tp;
        tp = Yc; Yc = Yn; Yn = tp;
        tp = Zc; Zc = Zn; Zn = tp;
    }

    // 4) GtG = G^T G  (G is 32x128)
    gemm(GtG, G, G, NXc, NXc, NYc, NXc, NXc, NXc, 1.0f, 0.0f, 1, nullptr);
    // 5) M1 = -0.5*(Q + GtG + eps I) + S
    k_build_m1<<<(NXc * NXc + 255) / 256, 256, 0, stream>>>(M1, Q, GtG, S);
    // 6) A = M1 @ P ; C = G @ P ; B = (sqrt(s)/denom) * Yfinal @ H
    gemm(Amat, M1, P, NXc, NXc, NXc, NXc, NXc, NXc, 1.0f, 0.0f, 0, nullptr);
    gemm(Cmat, G, P, NYc, NXc, NXc, NXc, NXc, NXc, 1.0f, 0.0f, 0, nullptr);
    gemm(Bmat, Yc, H, NXc, NUc, NXc, NXc, NUc, NUc, 1.0f, 0.0f, 0, scal + 2);

    // 7) pack Wc into WMMA B-operand layout
    int npack = (COLS / 16) * 48 * 32;
    k_pack<<<(npack + 255) / 256, 256, 0, stream>>>(Wpack, Amat, Bmat, Cmat);

    // 8) big fused GEMM with FP32 WMMA (64 samples per block)
    k_main<<<(int)(NN / (RT * 16)), 320, 0, stream>>>(x, u, Wpack, out, NN);
}